// RerankNet_36799279792399
// MI455X (gfx1250) — compile-verified
//
#include <hip/hip_runtime.h>
#include <hip/hip_bf16.h>

// ---------------------------------------------------------------------------
// RerankNet for MI455X (gfx1250, wave32).
//   in: hidden (1024x128 f32), logits (1024x100000 f32), emb (100000x128 f32),
//       W (3x128x128 f32), bias (3x128 f32)
//   out: logits with top-1000 positions replaced by reranked scores (f32)
// ---------------------------------------------------------------------------

typedef __attribute__((ext_vector_type(2))) float v2f;
typedef __attribute__((ext_vector_type(8))) float v8f;

#define BATCH   1024
#define VOCAB   100000
#define DIM     128
#define NCUT    3

// ---------------------------------------------------------------------------
// Kernel 1: R[i] = hidden @ W[i]^T + b[i]   via V_WMMA_F32_16X16X4_F32.
// One wave (32 lanes) computes one 16x16 output tile, looping K in steps of 4.
// A frag (16x4 f32, 2 VGPR): lanes 0-15 = M rows, v0/v1 hold K={0,1} (lo half)
// or K={2,3} (hi half). B frag (4x16): same K striping, N across lanes.
// B[k][n] = W[n][k] (because the GEMM is against W^T).
// ---------------------------------------------------------------------------
__global__ void rerank_gemm_wmma(const float* __restrict__ H,
                                 const float* __restrict__ W,
                                 const float* __restrict__ bias,
                                 float* __restrict__ R) {
    const int lane = threadIdx.x;          // 0..31
    const int lo   = lane & 15;
    const int hi   = lane >> 4;            // 0 or 1
    const int tm   = blockIdx.x * 16;      // M tile (1024/16 = 64)
    const int tn   = blockIdx.y * 16;      // N tile (128/16  = 8)
    const int cut  = blockIdx.z;           // 0..2

    const float* __restrict__ Wi = W + (size_t)cut * DIM * DIM;
    const float* __restrict__ Ha = H + (size_t)(tm + lo) * DIM;
    const float* __restrict__ Wn = Wi + (size_t)(tn + lo) * DIM;

    v8f acc = {};
#pragma unroll 4
    for (int k0 = 0; k0 < DIM; k0 += 4) {
        const int ka = k0 + 2 * hi;
        v2f a, b;
        a.x = Ha[ka];
        a.y = Ha[ka + 1];
        b.x = Wn[ka];        // B[ka][tn+lo]   = W[tn+lo][ka]
        b.y = Wn[ka + 1];    // B[ka+1][tn+lo] = W[tn+lo][ka+1]
        acc = __builtin_amdgcn_wmma_f32_16x16x4_f32(
                  /*neg_a=*/false, a, /*neg_b=*/false, b,
                  /*c_mod=*/(short)0, acc, /*reuse_a=*/false, /*reuse_b=*/false);
    }

    const float bn = bias[cut * DIM + tn + lo];
#pragma unroll
    for (int r = 0; r < 8; ++r) {
        const int row = tm + r + 8 * hi;   // C/D layout: vgpr r -> M=r (+8 hi)
        R[((size_t)cut * BATCH + row) * DIM + tn + lo] = acc[r] + bn;
    }
}

// ---------------------------------------------------------------------------
// Kernel 2: per-row radix select of the rank-{100,500,1000} threshold keys.
// One 256-thread block per row; dynamic LDS: 64K-bin u32 histogram (256 KB,
// within the 320 KB WGP LDS) + 256 group sums + select scratch.
// Order key: key = bits ^ (sign ? 0xFFFFFFFF : 0x80000000)  (larger f -> larger key)
// ---------------------------------------------------------------------------
__device__ __forceinline__ unsigned order_key(float x) {
    unsigned u = __float_as_uint(x);
    return u ^ ((u >> 31) ? 0xFFFFFFFFu : 0x80000000u);
}

// Single-thread: find bin holding the k-th largest (1-indexed) entry of hist,
// using 256-wide group sums to keep the serial scan ~512 iterations.
__device__ __forceinline__ void find_kth(const unsigned* hist, const unsigned* gsum,
                                         unsigned k, unsigned* outBin, unsigned* outRem) {
    unsigned cum = 0;
    int g = 255;
    for (; g > 0; --g) {
        if (cum + gsum[g] >= k) break;
        cum += gsum[g];
    }
    int b = g * 256 + 255;
    for (; b > g * 256; --b) {
        if (cum + hist[b] >= k) break;
        cum += hist[b];
    }
    *outBin = (unsigned)b;
    *outRem = k - cum;   // rank within the bin, 1-indexed
}

__global__ void topk_thresholds(const float* __restrict__ logits,
                                unsigned* __restrict__ T) {
    extern __shared__ unsigned smem[];
    unsigned* hist = smem;                 // 65536 bins
    unsigned* gsum = smem + 65536;         // 256 group sums
    unsigned* sSel = smem + 65536 + 256;   // [0..2]=prefix bin, [3..5]=residual rank

    const int row = blockIdx.x;
    const int tid = threadIdx.x;           // 256 threads
    const float* __restrict__ L = logits + (size_t)row * VOCAB;

    // ---- pass 1: histogram of top 16 key bits ----
    for (int b = tid; b < 65536; b += 256) hist[b] = 0;
    __syncthreads();
    for (int v = tid; v < VOCAB; v += 256)
        atomicAdd(&hist[order_key(L[v]) >> 16], 1u);
    __syncthreads();
    {
        unsigned s = 0;
        const int base = tid * 256;
        for (int b = 0; b < 256; ++b) s += hist[base + b];
        gsum[tid] = s;
    }
    __syncthreads();
    if (tid == 0) {
        const unsigned ranks[NCUT] = {100u, 500u, 1000u};
        for (int i = 0; i < NCUT; ++i) {
            unsigned bin, rem;
            find_kth(hist, gsum, ranks[i], &bin, &rem);
            sSel[i] = bin;
            sSel[NCUT + i] = rem;
        }
    }
    __syncthreads();

    // ---- pass 2: resolve low 16 bits for each rank boundary ----
    for (int i = 0; i < NCUT; ++i) {
        const unsigned pref = sSel[i];
        const unsigned rem  = sSel[NCUT + i];
        __syncthreads();
        for (int b = tid; b < 65536; b += 256) hist[b] = 0;
        __syncthreads();
        for (int v = tid; v < VOCAB; v += 256) {
            unsigned key = order_key(L[v]);
            if ((key >> 16) == pref) atomicAdd(&hist[key & 0xFFFFu], 1u);
        }
        __syncthreads();
        {
            unsigned s = 0;
            const int base = tid * 256;
            for (int b = 0; b < 256; ++b) s += hist[base + b];
            gsum[tid] = s;
        }
        __syncthreads();
        if (tid == 0) {
            unsigned bin, r2;
            find_kth(hist, gsum, rem, &bin, &r2);
            T[row * NCUT + i] = (pref << 16) | bin;
        }
    }
}

// ---------------------------------------------------------------------------
// Kernel 3: fused copy + gather + dot + scatter. One block per row; single
// pass over the 100K columns. Selected columns (key >= t1000) get the dot of
// the segment's reranked hidden (LDS-resident) with the L2-resident embedding
// row; everything else is copied through.
// ---------------------------------------------------------------------------
__global__ void rerank_scatter(const float* __restrict__ logits,
                               const float* __restrict__ emb,
                               const float* __restrict__ R,
                               const unsigned* __restrict__ T,
                               float* __restrict__ out) {
    __shared__ __align__(16) float rh[NCUT][DIM];
    __shared__ unsigned tkey[NCUT];

    const int row = blockIdx.x;
    const int tid = threadIdx.x;   // 256 threads
    if (tid < NCUT * DIM) {
        const int s = tid / DIM, k = tid % DIM;
        rh[s][k] = R[((size_t)s * BATCH + row) * DIM + k];
    }
    if (tid < NCUT) tkey[tid] = T[row * NCUT + tid];
    __syncthreads();

    const float* __restrict__ L = logits + (size_t)row * VOCAB;
    float* __restrict__ O = out + (size_t)row * VOCAB;

    for (int v = tid; v < VOCAB; v += 256) {
        __builtin_prefetch(&L[v + 2048], 0, 1);   // global_prefetch_b8
        const float x = L[v];
        const unsigned key = order_key(x);
        float result = x;
        if (key >= tkey[2]) {
            const int seg = (key >= tkey[0]) ? 0 : (key >= tkey[1]) ? 1 : 2;
            const float4* __restrict__ e = (const float4*)(emb + (size_t)v * DIM);
            const float4* __restrict__ h = (const float4*)(&rh[seg][0]);
            float acc = 0.f;
#pragma unroll 8
            for (int q = 0; q < DIM / 4; ++q) {
                const float4 ev = e[q];
                const float4 hv = h[q];
                acc += ev.x * hv.x + ev.y * hv.y + ev.z * hv.z + ev.w * hv.w;
            }
            result = acc;
        }
        O[v] = result;
    }
}

// ---------------------------------------------------------------------------
extern "C" void kernel_launch(void* const* d_in, const int* in_sizes, int n_in,
                              void* d_out, int out_size, void* d_ws, size_t ws_size,
                              hipStream_t stream) {
    const float* H      = (const float*)d_in[0];   // 1024 x 128
    const float* logits = (const float*)d_in[1];   // 1024 x 100000
    const float* emb    = (const float*)d_in[2];   // 100000 x 128
    const float* W      = (const float*)d_in[3];   // 3 x 128 x 128
    const float* bias   = (const float*)d_in[4];   // 3 x 128
    float* out = (float*)d_out;

    float*    R = (float*)d_ws;                                        // 3*1024*128 f32
    unsigned* T = (unsigned*)((char*)d_ws +
                              (size_t)NCUT * BATCH * DIM * sizeof(float)); // 1024*3 u32

    // 1) rerank hidden states on the WMMA path
    dim3 gGemm(BATCH / 16, DIM / 16, NCUT);
    rerank_gemm_wmma<<<gGemm, 32, 0, stream>>>(H, W, bias, R);

    // 2) per-row rank-{100,500,1000} threshold keys (radix select in LDS)
    const size_t smemBytes = (65536u + 256u + 8u) * sizeof(unsigned);  // ~257 KB < 320 KB
    topk_thresholds<<<BATCH, 256, smemBytes, stream>>>(logits, T);

    // 3) fused copy + gather + dot + scatter (single pass over logits)
    rerank_scatter<<<BATCH, 256, 0, stream>>>(logits, emb, R, T, out);
}